// BaseDGCNNGFModule_37125697307420
// MI455X (gfx1250) — compile-verified
//
#include <hip/hip_runtime.h>
#include <hip/hip_bf16.h>

// Problem constants (from reference)
#define B_   8
#define N_   4096
#define C_   64
#define K_   16
#define CNT_ (B_ * N_ * K_)   // 524288 rows for BN statistics
#define EPS_ 1e-5f

typedef __attribute__((ext_vector_type(2))) float v2f;
typedef __attribute__((ext_vector_type(8))) float v8f;

// -----------------------------------------------------------------------------
// LDS byte-offset of a __shared__ object (generic -> AS(3) -> 32-bit offset)
// -----------------------------------------------------------------------------
__device__ inline unsigned lds_off(const void* p) {
  return (unsigned)(size_t)(__attribute__((address_space(3))) const void*)p;
}

// -----------------------------------------------------------------------------
// Async global->LDS 16-byte copy (CDNA5 GLOBAL_LOAD_ASYNC_TO_LDS_B128,
// tracked by ASYNCcnt). Per-lane LDS dest address in VDST, 64-bit global
// address in VADDR.
// -----------------------------------------------------------------------------
__device__ inline void async_g2l_b128(void* lds_dst, const void* gsrc) {
  unsigned l = lds_off(lds_dst);
  unsigned long long g = (unsigned long long)(size_t)gsrc;
  asm volatile("global_load_async_to_lds_b128 %0, %1, off"
               :: "v"(l), "v"(g)
               : "memory");
}
__device__ inline void wait_async_le8() {
  asm volatile("s_wait_asynccnt 0x8" ::: "memory");
}
__device__ inline void wait_async_0() {
  asm volatile("s_wait_asynccnt 0x0" ::: "memory");
}

// -----------------------------------------------------------------------------
// 16x16 f32 tile: D[m][n] = sum_c A[m][c]*Bt[n][c] via chained
// V_WMMA_F32_16X16X4_F32 over the 64-deep inner dimension.
// Per-lane operand fetch per ISA layout (A 16x4 f32): lane L -> row = L&15,
// VGPR0/1 hold K = 2*(L>>4) and 2*(L>>4)+1 within each 4-wide K chunk.
// -----------------------------------------------------------------------------
__device__ inline v8f tile_dot64(const float* A, int lda, const float* Bt, int ldb, v8f acc) {
  const int lane = threadIdx.x & 31;
  const int r  = lane & 15;
  const int ks = (lane >> 4) << 1;
  const float* ap = A  + r * lda + ks;
  const float* bp = Bt + r * ldb + ks;
#pragma unroll
  for (int kb = 0; kb < 64; kb += 4) {
    v2f a = { ap[kb], ap[kb + 1] };
    v2f b = { bp[kb], bp[kb + 1] };
    acc = __builtin_amdgcn_wmma_f32_16x16x4_f32(false, a, false, b, (short)0, acc, false, false);
  }
  return acc;
}

// -----------------------------------------------------------------------------
// Kernel 0: zero the BN statistics accumulators (sum1, sumsq1, sum2, sumsq2)
// -----------------------------------------------------------------------------
__global__ __launch_bounds__(256) void zero_stats_kernel(float* stats) {
  stats[threadIdx.x] = 0.0f;
}

// -----------------------------------------------------------------------------
// Kernel 1: squared norms sq[b*N+n] = sum_c p^2
// -----------------------------------------------------------------------------
__global__ __launch_bounds__(256) void sqnorm_kernel(const float* __restrict__ pts,
                                                     float* __restrict__ sq) {
  int g = blockIdx.x * 256 + threadIdx.x;          // b*N + n
  const float* p = pts + (size_t)g * C_;
  float s = 0.0f;
#pragma unroll 16
  for (int c = 0; c < C_; ++c) s += p[c] * p[c];
  sq[g] = s;
}

// -----------------------------------------------------------------------------
// Kernel 2: KNN. Block = 256 threads (8 waves), 32 query rows of one batch.
// Column tiles of 128 are streamed into a DOUBLE-BUFFERED LDS region with
// async global->LDS B128 loads (ASYNCcnt), overlapped with the WMMA Gram
// computation on the current tile. Ranking key: sq[col] - 2*dot (row norm
// dropped, it is constant per row). Per-thread running top-16 + LDS merge.
// -----------------------------------------------------------------------------
__global__ __launch_bounds__(256) void knn_kernel(const float* __restrict__ pts,
                                                  const float* __restrict__ sq,
                                                  int* __restrict__ idxout) {
  __shared__ float smem[22656];                 // 88.5 KB
  float* prow  = smem;                          // 32*64   = 2048
  float* pcol0 = smem + 2048;                   // 128*64  = 8192 (buffer 0)
  float* pcol1 = smem + 10240;                  // 128*64  = 8192 (buffer 1)
  float* sqcol = smem + 18432;                  // 128
  float* score = smem + 18560;                  // 32*128  = 4096
  float* mdist = pcol0;                         // 32*128 (alias, after col loop)
  int*   midx  = (int*)(pcol0 + 4096);          // 32*128 (alias)

  const int b    = blockIdx.y;
  const int m0   = blockIdx.x * 32;
  const int t    = threadIdx.x;
  const int wave = t >> 5;
  const int lane = t & 31;
  const float* pb = pts + (size_t)b * N_ * C_;

  for (int i = t; i < 32 * 64; i += 256) prow[i] = pb[(size_t)m0 * 64 + i];

  float bd[16]; int bi[16];
#pragma unroll
  for (int j = 0; j < 16; ++j) { bd[j] = 3.0e38f; bi[j] = 0; }
  float worst = 3.0e38f; int wslot = 0;

  const int NB = N_ / 128;
  // prime the pipeline: async-fill buffer 0 with column tile 0 (8 x B128/thread)
#pragma unroll
  for (int j = 0; j < 8; ++j) {
    const int i = t + j * 256;                  // 16-byte chunk index within tile
    async_g2l_b128(pcol0 + i * 4, pb + (size_t)i * 4);
  }

  for (int cb = 0; cb < NB; ++cb) {
    const int c0 = cb * 128;
    float* cur = (cb & 1) ? pcol1 : pcol0;
    if (cb + 1 < NB) {
      // issue next tile into the other buffer, then wait only for current tile
      float* nxt = (cb & 1) ? pcol0 : pcol1;
      const float* gsrc = pb + (size_t)(c0 + 128) * 64;
#pragma unroll
      for (int j = 0; j < 8; ++j) {
        const int i = t + j * 256;
        async_g2l_b128(nxt + i * 4, gsrc + (size_t)i * 4);
      }
      wait_async_le8();   // in-order completion => current tile's 8 loads done
    } else {
      wait_async_0();
    }
    if (t < 128) sqcol[t] = sq[b * N_ + c0 + t];
    __syncthreads();      // publish async-filled LDS across all waves

    // 2x8 grid of 16x16 tiles; this wave computes tiles (0,wave) and (1,wave)
#pragma unroll
    for (int tm = 0; tm < 2; ++tm) {
      v8f acc = {};
      acc = tile_dot64(prow + tm * 16 * 64, 64, cur + wave * 16 * 64, 64, acc);
      const int n   = wave * 16 + (lane & 15);
      const int mb  = tm * 16 + ((lane >> 4) << 3);
      const float q = sqcol[n];
#pragma unroll
      for (int g = 0; g < 8; ++g)
        score[(mb + g) * 128 + n] = q - 2.0f * acc[g];
    }
    __syncthreads();      // all pcol/sqcol reads done before next overwrite

    // top-k update: 8 threads per row, each scans 16 columns
    const int r = t >> 3, sub = t & 7;
#pragma unroll
    for (int j = 0; j < 16; ++j) {
      const float d = score[r * 128 + sub * 16 + j];
      if (d < worst) {
        bd[wslot] = d; bi[wslot] = c0 + sub * 16 + j;
        worst = bd[0]; wslot = 0;
#pragma unroll
        for (int q = 1; q < 16; ++q)
          if (bd[q] > worst) { worst = bd[q]; wslot = q; }
      }
    }
  }
  __syncthreads();

  // dump 8x16 candidates per row, then one thread per row selects 16 smallest
  {
    const int r = t >> 3, sub = t & 7;
#pragma unroll
    for (int j = 0; j < 16; ++j) {
      mdist[r * 128 + sub * 16 + j] = bd[j];
      midx [r * 128 + sub * 16 + j] = bi[j];
    }
  }
  __syncthreads();
  if (t < 32) {
    float* dd = mdist + t * 128;
    int*   ii = midx  + t * 128;
    int*   og = idxout + ((size_t)b * N_ + m0 + t) * K_;
    for (int s = 0; s < K_; ++s) {
      float mn = dd[0]; int mj = 0;
      for (int q = 1; q < 128; ++q)
        if (dd[q] < mn) { mn = dd[q]; mj = q; }
      og[s] = ii[mj];
      dd[mj] = 3.0e38f;
    }
  }
}

// -----------------------------------------------------------------------------
// Kernel 3: gather neighbors + GEMM1 + BN1 statistics.
// edge@W1^T decomposed: z1 = grouped@W1a^T + center@(W1b - W1a)^T.
// Block: 4 points (64 rows of A = 4n x 16k), full 64-ch output, WMMA tiles.
// -----------------------------------------------------------------------------
__global__ __launch_bounds__(256) void edge_gemm1_kernel(const float* __restrict__ pts,
                                                         const float* __restrict__ W1,
                                                         const int* __restrict__ nidx,
                                                         float* __restrict__ z,
                                                         float* __restrict__ sum1,
                                                         float* __restrict__ sumsq1) {
  __shared__ float amat[64 * 64];    // gathered neighbor features
  __shared__ float w1s[64 * 128];    // full W1 (row-major, ld=128)
  __shared__ float cpts[4 * 64];
  __shared__ float cterm[4 * 64];
  __shared__ float ssum[64], ssq[64];

  const int b    = blockIdx.y;
  const int n0   = blockIdx.x * 4;
  const int t    = threadIdx.x;
  const int wave = t >> 5;
  const int lane = t & 31;
  const float* pb = pts + (size_t)b * N_ * C_;

  for (int i = t; i < 64 * 128; i += 256) w1s[i] = W1[i];
  if (t < 64) { ssum[t] = 0.0f; ssq[t] = 0.0f; }
  for (int i = t; i < 4 * 64; i += 256) cpts[i] = pb[(size_t)n0 * 64 + i];

  {  // gather: 64 rows, 4 threads/row, 16 floats each
    const int row = t >> 2;
    const int seg = (t & 3) * 16;
    const int nl = row >> 4, k = row & 15;
    const int gi = nidx[((size_t)b * N_ + n0 + nl) * K_ + k];
    const float* src = pb + (size_t)gi * 64 + seg;
    float* dst = amat + row * 64 + seg;
#pragma unroll
    for (int j = 0; j < 16; ++j) dst[j] = src[j];
  }
  __syncthreads();

  {  // center term: one (nl, o) pair per thread
    const int nl = t >> 6;
    const int o  = t & 63;
    const float* wrow = w1s + o * 128;
    const float* p = cpts + nl * 64;
    float s = 0.0f;
#pragma unroll 16
    for (int c = 0; c < 64; ++c) s += p[c] * (wrow[64 + c] - wrow[c]);
    cterm[nl * 64 + o] = s;
  }
  __syncthreads();

  // 4x4 grid of 16x16 output tiles; each wave does 2
  for (int tt = wave; tt < 16; tt += 8) {
    const int tm = tt >> 2, tn = tt & 3;
    v8f acc = {};
    acc = tile_dot64(amat + tm * 16 * 64, 64, w1s + tn * 16 * 128, 128, acc);
    const int o  = tn * 16 + (lane & 15);
    const int mb = tm * 16 + ((lane >> 4) << 3);
    float ls = 0.0f, lq = 0.0f;
#pragma unroll
    for (int g = 0; g < 8; ++g) {
      const int row = mb + g;
      const int nl = row >> 4, k = row & 15;
      const float v = acc[g] + cterm[nl * 64 + o];
      z[(((size_t)b * N_ + n0 + nl) * K_ + k) * 64 + o] = v;
      ls += v; lq += v * v;
    }
    atomicAdd(&ssum[o], ls);
    atomicAdd(&ssq[o], lq);
  }
  __syncthreads();
  if (t < 64) { atomicAdd(&sum1[t], ssum[t]); atomicAdd(&sumsq1[t], ssq[t]); }
}

// -----------------------------------------------------------------------------
// Kernel 4: BN1+ReLU applied while loading, GEMM2 (in-place on z), BN2 stats.
// -----------------------------------------------------------------------------
__global__ __launch_bounds__(256) void bn_gemm2_kernel(const float* __restrict__ W2,
                                                       float* __restrict__ z,
                                                       const float* __restrict__ sum1,
                                                       const float* __restrict__ sumsq1,
                                                       const float* __restrict__ gamma1,
                                                       const float* __restrict__ beta1,
                                                       float* __restrict__ sum2,
                                                       float* __restrict__ sumsq2) {
  __shared__ float amat[64 * 64];
  __shared__ float w2s[64 * 64];
  __shared__ float sc1[64], sh1[64];
  __shared__ float ssum[64], ssq[64];

  const int b    = blockIdx.y;
  const int n0   = blockIdx.x * 4;
  const int t    = threadIdx.x;
  const int wave = t >> 5;
  const int lane = t & 31;

  for (int i = t; i < 64 * 64; i += 256) w2s[i] = W2[i];
  if (t < 64) {
    const float m  = sum1[t] * (1.0f / CNT_);
    const float vr = sumsq1[t] * (1.0f / CNT_) - m * m;
    const float rs = rsqrtf(vr + EPS_);
    sc1[t] = rs * gamma1[t];
    sh1[t] = beta1[t] - m * rs * gamma1[t];
    ssum[t] = 0.0f; ssq[t] = 0.0f;
  }
  __syncthreads();

  const size_t zbase = ((size_t)b * N_ + n0) * K_ * 64;  // contiguous 64x64 block
  {  // load rows, apply BN1 + ReLU
    const int row = t >> 2;
    const int seg = (t & 3) * 16;
    const float* src = z + zbase + (size_t)row * 64 + seg;
    float* dst = amat + row * 64 + seg;
#pragma unroll
    for (int j = 0; j < 16; ++j) {
      const int c = seg + j;
      float v = src[j] * sc1[c] + sh1[c];
      dst[j] = v > 0.0f ? v : 0.0f;
    }
  }
  __syncthreads();

  for (int tt = wave; tt < 16; tt += 8) {
    const int tm = tt >> 2, tn = tt & 3;
    v8f acc = {};
    acc = tile_dot64(amat + tm * 16 * 64, 64, w2s + tn * 16 * 64, 64, acc);
    const int o  = tn * 16 + (lane & 15);
    const int mb = tm * 16 + ((lane >> 4) << 3);
    float ls = 0.0f, lq = 0.0f;
#pragma unroll
    for (int g = 0; g < 8; ++g) {
      const float v = acc[g];
      z[zbase + (size_t)(mb + g) * 64 + o] = v;   // in-place: block owns these rows
      ls += v; lq += v * v;
    }
    atomicAdd(&ssum[o], ls);
    atomicAdd(&ssq[o], lq);
  }
  __syncthreads();
  if (t < 64) { atomicAdd(&sum2[t], ssum[t]); atomicAdd(&sumsq2[t], ssq[t]); }
}

// -----------------------------------------------------------------------------
// Kernel 5: BN2 + ReLU + max over K. One thread per (b,n,o).
// -----------------------------------------------------------------------------
__global__ __launch_bounds__(256) void bn_maxk_kernel(const float* __restrict__ z,
                                                      const float* __restrict__ sum2,
                                                      const float* __restrict__ sumsq2,
                                                      const float* __restrict__ gamma2,
                                                      const float* __restrict__ beta2,
                                                      float* __restrict__ out) {
  const int g = blockIdx.x * 256 + threadIdx.x;   // (b*N + n)*64 + o
  const int o = g & 63;
  const size_t base = (size_t)(g >> 6) * (K_ * 64) + o;
  const float m  = sum2[o] * (1.0f / CNT_);
  const float vr = sumsq2[o] * (1.0f / CNT_) - m * m;
  const float rs = rsqrtf(vr + EPS_);
  const float sc = rs * gamma2[o];
  const float sh = beta2[o] - m * rs * gamma2[o];
  float mx = 0.0f;                                // relu() >= 0
#pragma unroll
  for (int k = 0; k < K_; ++k) {
    float v = z[base + (size_t)k * 64] * sc + sh;
    v = v > 0.0f ? v : 0.0f;
    mx = v > mx ? v : mx;
  }
  out[g] = mx;
}

// -----------------------------------------------------------------------------
// Host launcher
// -----------------------------------------------------------------------------
extern "C" void kernel_launch(void* const* d_in, const int* in_sizes, int n_in,
                              void* d_out, int out_size, void* d_ws, size_t ws_size,
                              hipStream_t stream) {
  const float* points = (const float*)d_in[0];
  const float* W1     = (const float*)d_in[1];
  const float* gamma1 = (const float*)d_in[2];
  const float* beta1  = (const float*)d_in[3];
  const float* W2     = (const float*)d_in[4];
  const float* gamma2 = (const float*)d_in[5];
  const float* beta2  = (const float*)d_in[6];
  float* out = (float*)d_out;

  char* ws = (char*)d_ws;
  float* sq    = (float*)ws;                                   // 128 KB
  float* stats = (float*)(ws + 131072);                        // 256 floats (1 KB, padded)
  int*   nidx  = (int*)  (ws + 131072 + 4096);                 // 2 MB
  float* z     = (float*)(ws + 131072 + 4096 + 2097152 + 4096);// 128 MB
  float* sum1 = stats, *sumsq1 = stats + 64, *sum2 = stats + 128, *sumsq2 = stats + 192;

  zero_stats_kernel<<<1, 256, 0, stream>>>(stats);
  sqnorm_kernel<<<(B_ * N_) / 256, 256, 0, stream>>>(points, sq);
  knn_kernel<<<dim3(N_ / 32, B_), 256, 0, stream>>>(points, sq, nidx);
  edge_gemm1_kernel<<<dim3(N_ / 4, B_), 256, 0, stream>>>(points, W1, nidx, z, sum1, sumsq1);
  bn_gemm2_kernel<<<dim3(N_ / 4, B_), 256, 0, stream>>>(W2, z, sum1, sumsq1, gamma1, beta1,
                                                        sum2, sumsq2);
  bn_maxk_kernel<<<(B_ * N_ * 64) / 256, 256, 0, stream>>>(z, sum2, sumsq2, gamma2, beta2, out);
}